// Net_16406775071044
// MI455X (gfx1250) — compile-verified
//
#include <hip/hip_runtime.h>
#include <hip/hip_bf16.h>

typedef __attribute__((ext_vector_type(2))) float v2f;
typedef __attribute__((ext_vector_type(8))) float v8f;

#define N_NODES 100000
#define N_EDGES 1000000
#define N_LABEL 200000
#define D 128
#define APITCH 132  // LDS row pitch (floats) to avoid bank conflicts

// ---------------------------------------------------------------------------
// Degree / normalization
// ---------------------------------------------------------------------------
__global__ void deg_init_kernel(float* __restrict__ deg, int n) {
    int i = blockIdx.x * blockDim.x + threadIdx.x;
    if (i < n) deg[i] = 1.0f;  // self-loop contributes 1
}

__global__ void deg_scatter_kernel(const long long* __restrict__ dstIdx,
                                   float* __restrict__ deg, int e) {
    int i = blockIdx.x * blockDim.x + threadIdx.x;
    if (i < e) unsafeAtomicAdd(&deg[(int)dstIdx[i]], 1.0f);
}

__global__ void deg_rsqrt_kernel(float* __restrict__ deg, int n) {
    int i = blockIdx.x * blockDim.x + threadIdx.x;
    if (i < n) deg[i] = rsqrtf(deg[i]);  // deg >= 1 always (self-loops)
}

// ---------------------------------------------------------------------------
// H = X @ W  via V_WMMA_F32_16X16X4_F32.
// Block = 256 threads = 8 waves. Block handles 16 rows; wave w handles cols
// [16w, 16w+16). K-loop: 32 x wmma 16x16x4. A tile staged in LDS.
// ---------------------------------------------------------------------------
__global__ __launch_bounds__(256) void gemm_wmma_kernel(
    const float* __restrict__ X, const float* __restrict__ W,
    float* __restrict__ H) {
    __shared__ float Atile[16 * APITCH];

    const int tid = threadIdx.x;
    const int m0  = blockIdx.x * 16;

    // Cooperative load of 16x128 A tile: thread -> (row = tid>>4, 8 floats)
    {
        const int row = tid >> 4;
        const int seg = tid & 15;
        const float4* s4 =
            reinterpret_cast<const float4*>(X + (size_t)(m0 + row) * D + seg * 8);
        float4* d4 = reinterpret_cast<float4*>(&Atile[row * APITCH + seg * 8]);
        d4[0] = s4[0];
        d4[1] = s4[1];
    }
    __syncthreads();

    const int wave = tid >> 5;        // 0..7 -> column tile
    const int lane = tid & 31;
    const int half = lane >> 4;       // 0 or 1
    const int l    = lane & 15;
    const int n0   = wave * 16;

    v8f c = {};
#pragma unroll
    for (int k0 = 0; k0 < D; k0 += 4) {
        // A 16x4: lane l holds row l, K = k0 + 2*half + {0,1}
        v2f a, b;
        a.x = Atile[l * APITCH + k0 + 2 * half + 0];
        a.y = Atile[l * APITCH + k0 + 2 * half + 1];
        // B 4x16: lane l holds col n0+l, K = k0 + 2*half + {0,1}
        b.x = W[(k0 + 2 * half + 0) * D + n0 + l];
        b.y = W[(k0 + 2 * half + 1) * D + n0 + l];
        c = __builtin_amdgcn_wmma_f32_16x16x4_f32(
            /*neg_a=*/false, a, /*neg_b=*/false, b,
            /*c_mod=*/(short)0, c, /*reuse_a=*/false, /*reuse_b=*/false);
    }

    // D tile: VGPR r -> row (r + 8*half), col (n0 + l)
#pragma unroll
    for (int r = 0; r < 8; ++r) {
        H[(size_t)(m0 + r + 8 * half) * D + n0 + l] = c[r];
    }
}

// ---------------------------------------------------------------------------
// AGG[i,:] = H[i,:] * dinv[i]^2   (self-loop term; also initializes buffer)
// ---------------------------------------------------------------------------
__global__ void agg_init_kernel(const float* __restrict__ H,
                                const float* __restrict__ dinv,
                                float* __restrict__ AGG) {
    int i = blockIdx.x * blockDim.x + threadIdx.x;  // over N_NODES*D
    if (i < N_NODES * D) {
        float w = dinv[i >> 7];
        AGG[i] = H[i] * (w * w);
    }
}

// ---------------------------------------------------------------------------
// AGG[dst,:] += H[src,:] * dinv[src]*dinv[dst].  One wave per edge,
// each lane handles 4 features (float4 gather + 4 no-return f32 atomics).
// ---------------------------------------------------------------------------
__global__ __launch_bounds__(256) void agg_scatter_kernel(
    const float* __restrict__ H, const float* __restrict__ dinv,
    const long long* __restrict__ srcIdx, const long long* __restrict__ dstIdx,
    float* __restrict__ AGG, int E) {
    int e = blockIdx.x * 8 + (threadIdx.x >> 5);
    int lane = threadIdx.x & 31;
    if (e >= E) return;
    int s = (int)srcIdx[e];
    int d = (int)dstIdx[e];
    float w = dinv[s] * dinv[d];
    const float4 hv =
        *reinterpret_cast<const float4*>(H + (size_t)s * D + lane * 4);
    float* base = AGG + (size_t)d * D + lane * 4;
    unsafeAtomicAdd(base + 0, hv.x * w);
    unsafeAtomicAdd(base + 1, hv.y * w);
    unsafeAtomicAdd(base + 2, hv.z * w);
    unsafeAtomicAdd(base + 3, hv.w * w);
}

// ---------------------------------------------------------------------------
// X[i,f] = (relu?) (X[i,f] + b[f])
// ---------------------------------------------------------------------------
__global__ void bias_act_kernel(float* __restrict__ X,
                                const float* __restrict__ b, int doRelu) {
    int i = blockIdx.x * blockDim.x + threadIdx.x;
    if (i < N_NODES * D) {
        float v = X[i] + b[i & (D - 1)];
        X[i] = doRelu ? fmaxf(v, 0.0f) : v;
    }
}

// ---------------------------------------------------------------------------
// pred[e] = dot(X[u], X[v]); one wave per label edge, shfl_xor reduction.
// ---------------------------------------------------------------------------
__global__ __launch_bounds__(256) void edge_score_kernel(
    const float* __restrict__ X, const long long* __restrict__ eli,
    float* __restrict__ out, int L) {
    int e = blockIdx.x * 8 + (threadIdx.x >> 5);
    int lane = threadIdx.x & 31;
    if (e >= L) return;
    int u = (int)eli[e];
    int v = (int)eli[(size_t)L + e];
    float4 a = *reinterpret_cast<const float4*>(X + (size_t)u * D + lane * 4);
    float4 b = *reinterpret_cast<const float4*>(X + (size_t)v * D + lane * 4);
    float p = a.x * b.x + a.y * b.y + a.z * b.z + a.w * b.w;
#pragma unroll
    for (int off = 16; off > 0; off >>= 1) p += __shfl_xor(p, off, 32);
    if (lane == 0) out[e] = p;
}

// ---------------------------------------------------------------------------
extern "C" void kernel_launch(void* const* d_in, const int* in_sizes, int n_in,
                              void* d_out, int out_size, void* d_ws, size_t ws_size,
                              hipStream_t stream) {
    const float*     X   = (const float*)d_in[0];
    const long long* ei  = (const long long*)d_in[1];
    const long long* eli = (const long long*)d_in[2];
    const float*     W1  = (const float*)d_in[3];
    const float*     b1  = (const float*)d_in[4];
    const float*     W2  = (const float*)d_in[5];
    const float*     b2  = (const float*)d_in[6];
    float*           out = (float*)d_out;

    // Workspace: dinv (N f32) | bufA (N*D f32) | bufB (N*D f32)  ~103 MB
    char*  ws   = (char*)d_ws;
    float* dinv = (float*)ws;
    size_t off  = ((size_t)N_NODES * 4 + 255) & ~(size_t)255;
    float* bufA = (float*)(ws + off);
    float* bufB = (float*)(ws + off + (size_t)N_NODES * D * sizeof(float));

    const long long* src = ei;            // edge_index[0]
    const long long* dst = ei + N_EDGES;  // edge_index[1]

    const int T = 256;
    const int gN   = (N_NODES + T - 1) / T;
    const int gE   = (N_EDGES + T - 1) / T;
    const int gND  = (N_NODES * D + T - 1) / T;
    const int gEsc = (N_EDGES + 7) / 8;
    const int gLbl = (N_LABEL + 7) / 8;
    const int gGemm = N_NODES / 16;  // 100000 = 6250*16 exactly

    // normalization
    deg_init_kernel<<<gN, T, 0, stream>>>(dinv, N_NODES);
    deg_scatter_kernel<<<gE, T, 0, stream>>>(dst, dinv, N_EDGES);
    deg_rsqrt_kernel<<<gN, T, 0, stream>>>(dinv, N_NODES);

    // layer 1: bufA = X@W1 ; bufB = aggregate ; relu+bias in place
    gemm_wmma_kernel<<<gGemm, T, 0, stream>>>(X, W1, bufA);
    agg_init_kernel<<<gND, T, 0, stream>>>(bufA, dinv, bufB);
    agg_scatter_kernel<<<gEsc, T, 0, stream>>>(bufA, dinv, src, dst, bufB, N_EDGES);
    bias_act_kernel<<<gND, T, 0, stream>>>(bufB, b1, 1);

    // layer 2: bufA = bufB@W2 ; bufB = aggregate ; bias in place
    gemm_wmma_kernel<<<gGemm, T, 0, stream>>>(bufB, W2, bufA);
    agg_init_kernel<<<gND, T, 0, stream>>>(bufA, dinv, bufB);
    agg_scatter_kernel<<<gEsc, T, 0, stream>>>(bufA, dinv, src, dst, bufB, N_EDGES);
    bias_act_kernel<<<gND, T, 0, stream>>>(bufB, b2, 0);

    // edge scoring
    edge_score_kernel<<<gLbl, T, 0, stream>>>(bufB, eli, out, N_LABEL);
}